// GNNstack_22308060135655
// MI455X (gfx1250) — compile-verified
//
#include <hip/hip_runtime.h>
#include <hip/hip_bf16.h>

typedef __attribute__((ext_vector_type(2))) float v2f;
typedef __attribute__((ext_vector_type(8))) float v8f;

// ---------------------------------------------------------------------------
// Degree / normalization: deg = 1 (self loop) + #in-edges; dinv = rsqrt(deg)
// ---------------------------------------------------------------------------
__global__ void deg_init_kernel(float* __restrict__ deg, int n) {
  int i = blockIdx.x * blockDim.x + threadIdx.x;
  if (i < n) deg[i] = 1.0f;  // self loop contribution
}

__global__ void deg_accum_kernel(const int* __restrict__ ei, float* __restrict__ deg, int e) {
  int i = blockIdx.x * blockDim.x + threadIdx.x;
  if (i < e) atomicAdd(&deg[ei[e + i]], 1.0f);  // ei row 1 = dst
}

__global__ void dinv_kernel(float* __restrict__ deg, int n) {
  int i = blockIdx.x * blockDim.x + threadIdx.x;
  if (i < n) deg[i] = rsqrtf(fmaxf(deg[i], 1e-12f));
}

// ---------------------------------------------------------------------------
// GEMM: Out[N x NCOLS] = A[N x 128] * W[128 x NCOLS] (+bias)(+sigmoid)
// 256 threads = 8 wave32; each wave owns 16 rows.
// Uses V_WMMA_F32_16X16X4_F32 (full fp32 matrix core path).
//
// W is staged in LDS PRE-PAIRED along K:
//   WL[kp * NCOLS + col] = { W[2kp][col], W[2kp+1][col] }   (float2)
// so each B fragment is a single even-aligned ds_load_b64 (no re-pack movs).
//
// A frag (16x4 fp32): lanes 0-15 rows, VGPR0/1 = K{0,1} (lanes<16) or K{2,3}.
// B frag (4x16 fp32): lanes 0-15 cols, VGPR0/1 = K{0,1} (lanes<16) or K{2,3}.
// C/D (16x16 fp32):   acc[r] = row (kh*8 + r), col = lane&15.
// ---------------------------------------------------------------------------
template <int NCOLS, int EPI>
__global__ __launch_bounds__(256) void gemm_wmma_kernel(
    const float* __restrict__ A, const float* __restrict__ W,
    const float* __restrict__ bias, float* __restrict__ Out, int n) {
  __shared__ v2f WL[64 * NCOLS];  // 128*NCOLS floats total
  {
    // Each thread covers 4 consecutive (kp,col) pairs via float4 reads of
    // rows 2kp and 2kp+1, interleaved into float2 LDS slots.
    constexpr int C4 = NCOLS / 4;
    for (int idx = threadIdx.x; idx < 64 * C4; idx += 256) {
      int kp = idx / C4;
      int c4 = idx % C4;
      float4 r0 = ((const float4*)(W + (size_t)(2 * kp) * NCOLS))[c4];
      float4 r1 = ((const float4*)(W + (size_t)(2 * kp + 1) * NCOLS))[c4];
      int base = kp * NCOLS + c4 * 4;
      WL[base + 0] = v2f{r0.x, r1.x};
      WL[base + 1] = v2f{r0.y, r1.y};
      WL[base + 2] = v2f{r0.z, r1.z};
      WL[base + 3] = v2f{r0.w, r1.w};
    }
  }
  __syncthreads();

  const int wave = threadIdx.x >> 5;
  const int lane = threadIdx.x & 31;
  const int row0 = blockIdx.x * 128 + wave * 16;
  const int m    = lane & 15;   // A row within tile == B/C/D column within tile
  const int kh   = lane >> 4;   // K-half selector for A/B frags, M-half for C/D
  const int ar   = min(row0 + m, n - 1);  // clamp so WMMA runs with full EXEC
  const float* Arow = A + (size_t)ar * 128;

  constexpr int NT = NCOLS / 16;
  v8f acc[NT] = {};

#pragma unroll 4
  for (int k0 = 0; k0 < 128; k0 += 4) {
    const int kk = k0 + kh * 2;
    v2f a;
    {
      const float2 av = *(const float2*)(Arow + kk);  // 8B-aligned (kk even)
      a.x = av.x;
      a.y = av.y;
    }
    const v2f* Brow = WL + ((k0 >> 1) + kh) * NCOLS + m;
#pragma unroll
    for (int t = 0; t < NT; ++t) {
      v2f b = Brow[t * 16];  // single ds_load_b64, imm offset t*128B
      acc[t] = __builtin_amdgcn_wmma_f32_16x16x4_f32(
          /*neg_a=*/false, a, /*neg_b=*/false, b,
          /*c_mod=*/(short)0, acc[t], /*reuse_a=*/false, /*reuse_b=*/false);
    }
  }

  const int mb = kh * 8;
#pragma unroll
  for (int t = 0; t < NT; ++t) {
#pragma unroll
    for (int r = 0; r < 8; ++r) {
      const int row = row0 + mb + r;
      if (row < n) {
        float v = acc[t][r];
        const int col = t * 16 + m;
        if (EPI >= 1) v += bias[col];
        if (EPI == 2) v = 1.0f / (1.0f + __expf(-v));
        Out[(size_t)row * NCOLS + col] = v;
      }
    }
  }
}

// ---------------------------------------------------------------------------
// Aggregation init: out = hw * dinv^2 (self loop) + bias  (also initializes)
// ---------------------------------------------------------------------------
__global__ void agg_init_kernel(const float* __restrict__ hw,
                                const float* __restrict__ dinv,
                                const float* __restrict__ bias,
                                float* __restrict__ out, int n) {
  int t = blockIdx.x * blockDim.x + threadIdx.x;
  if (t >= n * 128) return;
  int row = t >> 7, col = t & 127;
  float di = dinv[row];
  out[t] = hw[t] * di * di + bias[col];
}

// ---------------------------------------------------------------------------
// Edge scatter: out[dst] += hw[src] * dinv[src]*dinv[dst]; 4 threads/edge.
// ---------------------------------------------------------------------------
__global__ void agg_edges_kernel(const float* __restrict__ hw,
                                 const int* __restrict__ ei,
                                 const float* __restrict__ dinv,
                                 float* __restrict__ out, int e) {
  int tid = blockIdx.x * blockDim.x + threadIdx.x;
  int edge = tid >> 2;
  if (edge >= e) return;
  int part = tid & 3;
  int s = ei[edge];
  int d = ei[e + edge];
  float nrm = dinv[s] * dinv[d];
  const float4* sp = (const float4*)(hw + (size_t)s * 128) + part * 8;
  float* dp = out + (size_t)d * 128 + part * 32;
#pragma unroll
  for (int i = 0; i < 8; ++i) {
    float4 v = sp[i];
    atomicAdd(dp + i * 4 + 0, v.x * nrm);
    atomicAdd(dp + i * 4 + 1, v.y * nrm);
    atomicAdd(dp + i * 4 + 2, v.z * nrm);
    atomicAdd(dp + i * 4 + 3, v.w * nrm);
  }
}

// ---------------------------------------------------------------------------
// ReLU (+ optional LayerNorm). One wave32 per 128-wide row; shfl reductions.
// ---------------------------------------------------------------------------
__global__ __launch_bounds__(256) void relu_ln_kernel(float* __restrict__ h,
                                                      const float* __restrict__ g,
                                                      const float* __restrict__ be,
                                                      int n, int do_ln) {
  int wave = threadIdx.x >> 5, lane = threadIdx.x & 31;
  int row = blockIdx.x * 8 + wave;
  if (row >= n) return;
  float4* p = (float4*)(h + (size_t)row * 128) + lane;
  float4 v = *p;
  v.x = fmaxf(v.x, 0.0f);
  v.y = fmaxf(v.y, 0.0f);
  v.z = fmaxf(v.z, 0.0f);
  v.w = fmaxf(v.w, 0.0f);
  if (do_ln) {
    float s = v.x + v.y + v.z + v.w;
    for (int o = 16; o > 0; o >>= 1) s += __shfl_xor(s, o, 32);
    float mu = s * (1.0f / 128.0f);
    float dx = v.x - mu, dy = v.y - mu, dz = v.z - mu, dw = v.w - mu;
    float q = dx * dx + dy * dy + dz * dz + dw * dw;
    for (int o = 16; o > 0; o >>= 1) q += __shfl_xor(q, o, 32);
    float rstd = rsqrtf(q * (1.0f / 128.0f) + 1e-5f);
    const float4 gv = ((const float4*)g)[lane];
    const float4 bv = ((const float4*)be)[lane];
    v.x = gv.x * dx * rstd + bv.x;
    v.y = gv.y * dy * rstd + bv.y;
    v.z = gv.z * dz * rstd + bv.z;
    v.w = gv.w * dw * rstd + bv.w;
  }
  *p = v;
}

// ---------------------------------------------------------------------------
extern "C" void kernel_launch(void* const* d_in, const int* in_sizes, int n_in,
                              void* d_out, int out_size, void* d_ws, size_t ws_size,
                              hipStream_t stream) {
  const float* x   = (const float*)d_in[0];
  const int*   ei  = (const int*)d_in[1];
  const float* W1  = (const float*)d_in[2];
  const float* b1  = (const float*)d_in[3];
  const float* W2  = (const float*)d_in[4];
  const float* b2  = (const float*)d_in[5];
  const float* W3  = (const float*)d_in[6];
  const float* b3  = (const float*)d_in[7];
  const float* g1  = (const float*)d_in[8];
  const float* be1 = (const float*)d_in[9];
  const float* g2  = (const float*)d_in[10];
  const float* be2 = (const float*)d_in[11];
  const float* Wp1 = (const float*)d_in[12];
  const float* bp1 = (const float*)d_in[13];
  const float* Wp2 = (const float*)d_in[14];
  const float* bp2 = (const float*)d_in[15];
  float* out = (float*)d_out;

  const int n = in_sizes[0] / 128;  // N nodes
  const int e = in_sizes[1] / 2;    // E edges

  float* ws = (float*)d_ws;
  size_t npad = ((size_t)n + 255) & ~(size_t)255;
  float* dinv = ws;                        // N floats
  float* bufA = ws + npad;                 // N x 128
  float* bufB = bufA + (size_t)n * 128;    // N x 128

  dim3 blk(256);
  int gb_n    = (n + 255) / 256;
  int gb_e    = (e + 255) / 256;
  int gb_ne   = (n * 128 + 255) / 256;
  int gb_e4   = (e * 4 + 255) / 256;
  int gb_gemm = (n + 127) / 128;
  int gb_ln   = (n + 7) / 8;

  // normalization coefficients
  deg_init_kernel<<<gb_n, blk, 0, stream>>>(dinv, n);
  deg_accum_kernel<<<gb_e, blk, 0, stream>>>(ei, dinv, e);
  dinv_kernel<<<gb_n, blk, 0, stream>>>(dinv, n);

  // layer 1: LN(relu(agg(x @ W1) + b1))
  gemm_wmma_kernel<128, 0><<<gb_gemm, blk, 0, stream>>>(x, W1, nullptr, bufA, n);
  agg_init_kernel<<<gb_ne, blk, 0, stream>>>(bufA, dinv, b1, bufB, n);
  agg_edges_kernel<<<gb_e4, blk, 0, stream>>>(bufA, ei, dinv, bufB, e);
  relu_ln_kernel<<<gb_ln, blk, 0, stream>>>(bufB, g1, be1, n, 1);

  // layer 2
  gemm_wmma_kernel<128, 0><<<gb_gemm, blk, 0, stream>>>(bufB, W2, nullptr, bufA, n);
  agg_init_kernel<<<gb_ne, blk, 0, stream>>>(bufA, dinv, b2, bufB, n);
  agg_edges_kernel<<<gb_e4, blk, 0, stream>>>(bufA, ei, dinv, bufB, e);
  relu_ln_kernel<<<gb_ln, blk, 0, stream>>>(bufB, g2, be2, n, 1);

  // layer 3 (relu only)
  gemm_wmma_kernel<128, 0><<<gb_gemm, blk, 0, stream>>>(bufB, W3, nullptr, bufA, n);
  agg_init_kernel<<<gb_ne, blk, 0, stream>>>(bufA, dinv, b3, bufB, n);
  agg_edges_kernel<<<gb_e4, blk, 0, stream>>>(bufA, ei, dinv, bufB, e);
  relu_ln_kernel<<<gb_ln, blk, 0, stream>>>(bufB, nullptr, nullptr, n, 0);

  // projection head: (h @ Wp1 + bp1) @ Wp2 + bp2, sigmoid
  gemm_wmma_kernel<128, 1><<<gb_gemm, blk, 0, stream>>>(bufB, Wp1, bp1, bufA, n);
  gemm_wmma_kernel<64, 2><<<gb_gemm, blk, 0, stream>>>(bufA, Wp2, bp2, out, n);
}